// GatherVertical_40656160424518
// MI455X (gfx1250) — compile-verified
//
#include <hip/hip_runtime.h>

typedef __attribute__((ext_vector_type(2))) float v2f;
typedef __attribute__((ext_vector_type(8))) float v8f;

#define TPB 192  // 6 waves of 32; each wave owns 2 M-tiles (32 output channels)

__global__ __launch_bounds__(TPB)
void Conv9FullWmmaF32_kernel(const float* __restrict__ in,
                             const float* __restrict__ wt,
                             const float* __restrict__ bs,
                             float* __restrict__ out) {
  constexpr int CI = 192;     // input channels (K of GEMM)
  constexpr int CO = 192;     // output channels (M of GEMM)
  constexpr int NCOL = 81;    // y*9+x columns (N of GEMM), padded to 96
  constexpr int LROW = 224;   // dwords per c-pair row: 192 data + 32 pad (bank split)

  extern __shared__ float xs[];  // xs[(c>>1)*LROW + col*2 + (c&1)] ; 96*224 floats

  const int b = blockIdx.x;  // batch
  const int k = blockIdx.y;  // kernel tap
  const int tid = threadIdx.x;

  // ---- Stage zero-padded, pair-interleaved X into LDS ----
  for (int i = tid; i < (96 * 192) / 4; i += TPB) {
    int j = i << 2;
    int row = j / 192;
    int off = j - row * 192;
    *(float4*)(xs + row * LROW + off) = make_float4(0.f, 0.f, 0.f, 0.f);
  }
  __syncthreads();

  const float* inb = in + (size_t)b * (CI * NCOL);
  for (int i = tid; i < CI * NCOL; i += TPB) {
    int c = i / NCOL;
    int r = i - c * NCOL;
    int h = r / 9;
    int x = r - h * 9;
    int y = h - k + 4;                      // shifted row
    if ((unsigned)y < 9u) {
      xs[(c >> 1) * LROW + (y * 9 + x) * 2 + (c & 1)] = inb[i];
    }
  }
  __syncthreads();

  // ---- GEMM: D[o, col] = sum_c W[k][o, c] * X[c, col] ----
  const int wave = tid >> 5;
  const int lane = tid & 31;
  const int half = lane >> 4;   // 0: lanes 0-15, 1: lanes 16-31
  const int l16  = lane & 15;
  const int obase = wave * 32;  // 6 waves x 32 rows = 192 output channels

  // A fragments: lane holds W[k, obase(+16)+l16, c0 + 2*half + {0,1}] (contiguous)
  const float* Wk0 = wt + (size_t)k * CO * CI + (size_t)(obase + l16) * CI + 2 * half;
  const float* Wk1 = Wk0 + 16 * CI;

  // B fragment LDS indices per N-tile: lane holds X[c0 + 2*half + {0,1}, t*16 + l16]
  int ib[6];
#pragma unroll
  for (int t = 0; t < 6; ++t) ib[t] = half * LROW + (t * 16 + l16) * 2;

  v8f acc0[6] = {};
  v8f acc1[6] = {};

#pragma unroll 2
  for (int c0 = 0; c0 < CI; c0 += 4) {
    v2f a0 = *(const v2f*)(Wk0 + c0);
    v2f a1 = *(const v2f*)(Wk1 + c0);
    const float* xr = xs + (c0 >> 1) * LROW;
    v2f bb[6];
#pragma unroll
    for (int t = 0; t < 6; ++t) bb[t] = *(const v2f*)(xr + ib[t]);
#pragma unroll
    for (int t = 0; t < 6; ++t) {
      acc0[t] = __builtin_amdgcn_wmma_f32_16x16x4_f32(
          false, a0, false, bb[t], (short)0, acc0[t], false, false);
      acc1[t] = __builtin_amdgcn_wmma_f32_16x16x4_f32(
          false, a1, false, bb[t], (short)0, acc1[t], false, false);
    }
  }

  // ---- Epilogue: bias add + non-temporal store of valid columns ----
  // C/D layout: vgpr r -> row obase + 16*m + r + 8*half, lane l16 -> column
#pragma unroll
  for (int m = 0; m < 2; ++m) {
    const v8f* acc = m ? acc1 : acc0;
    const int orow = obase + 16 * m + 8 * half;
    const float* bp = bs + k * CO + orow;   // 32B aligned (orow multiple of 8)
    float4 b0 = *(const float4*)(bp);
    float4 b1 = *(const float4*)(bp + 4);
    float biasr[8] = {b0.x, b0.y, b0.z, b0.w, b1.x, b1.y, b1.z, b1.w};

    float* op = out + ((size_t)(k * 512 + b) * CO + orow) * NCOL;
#pragma unroll
    for (int t = 0; t < 6; ++t) {
      int col = t * 16 + l16;
      if (col < NCOL) {
#pragma unroll
        for (int r = 0; r < 8; ++r) {
          __builtin_nontemporal_store(acc[t][r] + biasr[r],
                                      op + (size_t)r * NCOL + col);
        }
      }
    }
  }
}

extern "C" void kernel_launch(void* const* d_in, const int* in_sizes, int n_in,
                              void* d_out, int out_size, void* d_ws, size_t ws_size,
                              hipStream_t stream) {
  (void)in_sizes; (void)n_in; (void)out_size; (void)d_ws; (void)ws_size;
  const float* in = (const float*)d_in[0];   // (512,192,9,9)
  const float* wt = (const float*)d_in[1];   // (9,192,192)
  const float* bs = (const float*)d_in[2];   // (9,192)
  float* out = (float*)d_out;                // (9,512,192,9,9)

  dim3 grid(512, 9);
  dim3 block(TPB);
  size_t shmem = (size_t)96 * 224 * sizeof(float);  // 86016 B
  Conv9FullWmmaF32_kernel<<<grid, block, shmem, stream>>>(in, wt, bs, out);
}